// SpatioTemporalTransformer_42185168781910
// MI455X (gfx1250) — compile-verified
//
#include <hip/hip_runtime.h>
#include <math.h>

#define TT 6
#define NN 512
#define DD 128
#define HH 8
#define HDD 16
#define FINN 15
#define LLAY 3
#define SS (TT * NN)   // 3072
#define FFN 2048

typedef float v2f __attribute__((ext_vector_type(2)));
typedef float v8f __attribute__((ext_vector_type(8)));
typedef unsigned int u32x4 __attribute__((ext_vector_type(4)));
typedef unsigned int u32x8 __attribute__((ext_vector_type(8)));

// ---------- monotonic float<->uint for atomic segment-max ----------
__device__ __forceinline__ unsigned f2ord(float f) {
    unsigned u = __float_as_uint(f);
    return (u & 0x80000000u) ? ~u : (u | 0x80000000u);
}
__device__ __forceinline__ float ord2f(unsigned u) {
    return __uint_as_float((u & 0x80000000u) ? (u & 0x7fffffffu) : ~u);
}

// ---------- Tensor Data Mover: 2D tile (f32) global -> LDS ----------
// D# per CDNA5 ISA ch.8: Group0 = {count/flags, lds_addr, global_addr, type=2},
// Group1 = {data_size/pad, tensor dims, tile dims, dim0 stride}. 2-group form
// (VADDR2/3 = NULL) is enough for 2D tiles. pad_interval_code inserts one
// dword of LDS padding every 2^(code+1) dwords (3 -> every 16, 5 -> every 64),
// giving conflict-free LDS strides of 17/65 for free.
__device__ __forceinline__ void tdm_load_2d(unsigned lds_addr, const float* gptr,
                                            unsigned tile_x, unsigned tile_y,
                                            unsigned tensor_x, unsigned tensor_y,
                                            unsigned stride_x,
                                            unsigned pad_interval_code) {
    unsigned long long ga = (unsigned long long)gptr;
    u32x4 g0;
    g0[0] = 1u;                                   // count=1, user descriptor
    g0[1] = lds_addr;                             // LDS byte address
    g0[2] = (unsigned)ga;                         // global_addr[31:0]
    g0[3] = (unsigned)(ga >> 32) | (2u << 30);    // global_addr[56:32] | type=2
    u32x8 g1;
    g1[0] = (2u << 16)                            // data_size = 4 bytes
          | (1u << 20)                            // pad_enable
          | (pad_interval_code << 22);            // pad_amount code 0 = 1 dword
    g1[1] = (tensor_x & 0xFFFFu) << 16;           // tensor_dim0[15:0]
    g1[2] = (tensor_x >> 16) | ((tensor_y & 0xFFFFu) << 16);
    g1[3] = (tensor_y >> 16) | (tile_x << 16);    // tile_dim0
    g1[4] = tile_y;                               // tile_dim1 (tile_dim2 = 0)
    g1[5] = stride_x;                             // tensor_dim0_stride[31:0]
    g1[6] = 0u;                                   // stride hi, dim1_stride lo
    g1[7] = 0u;
    asm volatile("tensor_load_to_lds %0, %1" ::"s"(g0), "s"(g1) : "memory");
}

__device__ __forceinline__ unsigned lds_off32(const void* p) {
    return (unsigned)(unsigned long long)p;       // low 32 bits = LDS byte offset
}

// ---------------- GAT: xn @ W  ->  h[t,n,h,d] ----------------
__global__ void k_gat_lin(const float* __restrict__ x, const float* __restrict__ W,
                          float* __restrict__ hgat) {
    int idx = blockIdx.x * blockDim.x + threadIdx.x;
    if (idx >= TT * NN * HH * DD) return;
    int c  = idx & (HH * DD - 1);   // h*128+d
    int tn = idx >> 10;             // t*N+n
    const float* xr = x + (size_t)tn * FINN;
    float acc = 0.f;
#pragma unroll
    for (int f = 0; f < FINN; ++f) acc = fmaf(xr[f], W[f * HH * DD + c], acc);
    hgat[idx] = acc;
}

// ------------- per-(t,n,h) attention logits: (h*a).sum(-1) -------------
__global__ void k_gat_alpha(const float* __restrict__ hgat, const float* __restrict__ asrc,
                            const float* __restrict__ adst, float* __restrict__ as_out,
                            float* __restrict__ ad_out) {
    int wave = blockIdx.x * (blockDim.x >> 5) + (threadIdx.x >> 5);
    if (wave >= TT * NN * HH) return;   // wave-uniform exit
    int lane = threadIdx.x & 31;
    int h = wave & 7;
    const float* hp = hgat + (size_t)wave * DD;
    float a0 = 0.f, a1 = 0.f;
#pragma unroll
    for (int i = 0; i < 4; ++i) {
        int d = lane * 4 + i;
        float hv = hp[d];
        a0 = fmaf(hv, asrc[h * DD + d], a0);
        a1 = fmaf(hv, adst[h * DD + d], a1);
    }
#pragma unroll
    for (int m = 16; m >= 1; m >>= 1) {
        a0 += __shfl_xor(a0, m, 32);
        a1 += __shfl_xor(a1, m, 32);
    }
    if (lane == 0) { as_out[wave] = a0; ad_out[wave] = a1; }
}

// ------------- edge pass 1: leaky-relu score + segment max -------------
__global__ void k_edge_score(const int* __restrict__ ei, const float* __restrict__ as,
                             const float* __restrict__ ad, float* __restrict__ exbuf,
                             unsigned* __restrict__ mbuf, int E) {
    int EN = E + NN;
    int idx = blockIdx.x * blockDim.x + threadIdx.x;
    if (idx >= TT * EN) return;
    int t = idx / EN, e = idx - t * EN;
    int src = (e < E) ? ei[e] : (e - E);
    int dst = (e < E) ? ei[E + e] : (e - E);
    const float* asp = as + (size_t)(t * NN + src) * HH;
    const float* adp = ad + (size_t)(t * NN + dst) * HH;
    float* ep = exbuf + (size_t)idx * HH;
    unsigned* mp = mbuf + (size_t)(t * NN + dst) * HH;
#pragma unroll
    for (int h = 0; h < HH; ++h) {
        float v = asp[h] + adp[h];
        v = (v > 0.f) ? v : 0.2f * v;   // LeakyReLU(0.2)
        ep[h] = v;
        atomicMax(&mp[h], f2ord(v));
    }
}

// ------------- edge pass 2: exp(e - m[dst]) + segment sum -------------
__global__ void k_edge_exp(const int* __restrict__ ei, const unsigned* __restrict__ mbuf,
                           float* __restrict__ exbuf, float* __restrict__ ssum, int E) {
    int EN = E + NN;
    int idx = blockIdx.x * blockDim.x + threadIdx.x;
    if (idx >= TT * EN) return;
    int t = idx / EN, e = idx - t * EN;
    int dst = (e < E) ? ei[E + e] : (e - E);
    float* ep = exbuf + (size_t)idx * HH;
    const unsigned* mp = mbuf + (size_t)(t * NN + dst) * HH;
    float* sp = ssum + (size_t)(t * NN + dst) * HH;
#pragma unroll
    for (int h = 0; h < HH; ++h) {
        float ev = __expf(ep[h] - ord2f(mp[h]));
        ep[h] = ev;
        atomicAdd(&sp[h], ev);
    }
}

// ------------- edge pass 3: scatter sum of a*h[src], fused head-mean -------------
__global__ void k_edge_agg(const int* __restrict__ ei, const float* __restrict__ exbuf,
                           const float* __restrict__ ssum, const float* __restrict__ hgat,
                           float* __restrict__ gout, int E) {
    int EN = E + NN;
    int idx = blockIdx.x;
    int t = idx / EN, e = idx - t * EN;
    int src = (e < E) ? ei[e] : (e - E);
    int dst = (e < E) ? ei[E + e] : (e - E);
    __shared__ float coeff[HH];
    int tid = threadIdx.x;
    if (tid < HH)
        coeff[tid] = exbuf[(size_t)idx * HH + tid] / ssum[(size_t)(t * NN + dst) * HH + tid];
    __syncthreads();
    const float* hp = hgat + (size_t)(t * NN + src) * HH * DD + tid;
    float acc = 0.f;
#pragma unroll
    for (int h = 0; h < HH; ++h) acc = fmaf(coeff[h], hp[h * DD], acc);
    atomicAdd(&gout[(size_t)(t * NN + dst) * DD + tid], acc * (1.f / HH));
}

// ------------- LayerNorm over D=128 (one block per row) -------------
__global__ void k_ln(const float* in, const float* addv, int add_is_row,
                     const float* __restrict__ g, const float* __restrict__ b, float* out) {
    int s = blockIdx.x, d = threadIdx.x;
    __shared__ float red[DD];
    float v = in[(size_t)s * DD + d] + (add_is_row ? addv[d] : addv[(size_t)s * DD + d]);
    red[d] = v;
    __syncthreads();
    for (int off = 64; off > 0; off >>= 1) { if (d < off) red[d] += red[d + off]; __syncthreads(); }
    float mean = red[0] * (1.f / DD);
    __syncthreads();
    float c = v - mean;
    red[d] = c * c;
    __syncthreads();
    for (int off = 64; off > 0; off >>= 1) { if (d < off) red[d] += red[d + off]; __syncthreads(); }
    float var = red[0] * (1.f / DD);
    out[(size_t)s * DD + d] = c * rsqrtf(var + 1e-5f) * g[d] + b[d];
}

// ------- f32 WMMA GEMM, TDM-fed, double-buffered LDS: C = A@B + bias (+ReLU) -------
// Block 256 = 8 waves -> 64x64 C macro-tile; wave (wrow,wcol) owns 16x32.
// Wave 0 DMA-issues A(64x16) and B(16x64) chunks via tensor_load_to_lds with
// LDS padding -> strides 17/65 (bank-conflict-free), pipelined one chunk ahead,
// retired with s_wait_tensorcnt (per-wave, in-order TDM completion).
__global__ void k_gemm(const float* __restrict__ A, const float* __restrict__ B,
                       const float* __restrict__ bias, float* __restrict__ C,
                       int M, int Nc, int K, int relu) {
    __shared__ float ldsA[2][64 * 17];
    __shared__ float ldsB[2][16 * 65];
    int nbn = Nc >> 6;
    int bm = blockIdx.x / nbn, bn = blockIdx.x - bm * nbn;
    int m0 = bm * 64, n0 = bn * 64;
    int tid = threadIdx.x;
    int wv = tid >> 5, lane = tid & 31, half = lane >> 4, l16 = lane & 15;
    int wrow = wv >> 1, wcol = wv & 1;
    int nch = K >> 4;
    bool issuer = (wv == 0);   // wave-uniform

    unsigned la0 = lds_off32(&ldsA[0][0]), la1 = lds_off32(&ldsA[1][0]);
    unsigned lb0 = lds_off32(&ldsB[0][0]), lb1 = lds_off32(&ldsB[1][0]);

    if (issuer) {
        tdm_load_2d(la0, A + (size_t)m0 * K, 16, 64, (unsigned)K, (unsigned)M, (unsigned)K, 3);
        tdm_load_2d(lb0, B + n0, 64, 16, (unsigned)Nc, (unsigned)K, (unsigned)Nc, 5);
    }
    v8f c0 = {}, c1 = {};
    for (int ch = 0; ch < nch; ++ch) {
        int buf = ch & 1;
        if (issuer) {
            if (ch + 1 < nch) {
                int k1 = (ch + 1) << 4;
                tdm_load_2d(buf ? la0 : la1, A + (size_t)m0 * K + k1,
                            16, 64, (unsigned)K, (unsigned)M, (unsigned)K, 3);
                tdm_load_2d(buf ? lb0 : lb1, B + (size_t)k1 * Nc + n0,
                            64, 16, (unsigned)Nc, (unsigned)K, (unsigned)Nc, 5);
                __builtin_amdgcn_s_wait_tensorcnt(2);   // oldest chunk landed
            } else {
                __builtin_amdgcn_s_wait_tensorcnt(0);
            }
        }
        __syncthreads();    // publish chunk `ch` to all waves
        const float* At = &ldsA[buf][(wrow * 16 + l16) * 17];
        const float* Bt = &ldsB[buf][wcol * 32 + l16];
#pragma unroll
        for (int k = 0; k < 16; k += 4) {
            int ka = k + half * 2;
            v2f a, b0v, b1v;
            a.x = At[ka];
            a.y = At[ka + 1];
            b0v.x = Bt[ka * 65];
            b0v.y = Bt[(ka + 1) * 65];
            b1v.x = Bt[ka * 65 + 16];
            b1v.y = Bt[(ka + 1) * 65 + 16];
            c0 = __builtin_amdgcn_wmma_f32_16x16x4_f32(false, a, false, b0v, (short)0, c0,
                                                       false, false);
            c1 = __builtin_amdgcn_wmma_f32_16x16x4_f32(false, a, false, b1v, (short)0, c1,
                                                       false, false);
        }
        __syncthreads();    // chunk consumed before its buffer is re-filled
    }
    int colA = n0 + wcol * 32 + l16;
    int row0 = m0 + wrow * 16 + half * 8;
    float bv0 = bias[colA], bv1 = bias[colA + 16];
#pragma unroll
    for (int r = 0; r < 8; ++r) {
        float v0 = c0[r] + bv0, v1 = c1[r] + bv1;
        if (relu) { v0 = fmaxf(v0, 0.f); v1 = fmaxf(v1, 0.f); }
        C[(size_t)(row0 + r) * Nc + colA] = v0;
        C[(size_t)(row0 + r) * Nc + colA + 16] = v1;
    }
}

// ------------- flash attention, f32 WMMA, one wave per (head, 16-query block) ------
__global__ void k_attn(const float* __restrict__ qkv, float* __restrict__ o_out) {
    __shared__ float psh[4][16][17];   // per-wave P staging (C-layout -> A-layout)
    int w = threadIdx.x >> 5;
    int wave = blockIdx.x * 4 + w;     // grid sized exactly: H*(S/16) waves
    int qb = wave % (SS / 16);
    int hh = wave / (SS / 16);
    int lane = threadIdx.x & 31, half = lane >> 4, l16 = lane & 15;
    int s0 = qb * 16;

    v2f qa[4];
    const float* qp = qkv + (size_t)(s0 + l16) * (3 * DD) + hh * HDD;
#pragma unroll
    for (int j = 0; j < 4; ++j) { int kk = 4 * j + half * 2; qa[j].x = qp[kk]; qa[j].y = qp[kk + 1]; }

    v8f o = {};
    float mrow[8], srow[8];
#pragma unroll
    for (int r = 0; r < 8; ++r) { mrow[r] = -3.0e38f; srow[r] = 0.f; }
    const float scale = 0.25f;   // 1/sqrt(HD=16)

    for (int kb = 0; kb < SS / 16; ++kb) {
        int kbase = kb * 16;
        v8f sc = {};
        const float* kp = qkv + (size_t)(kbase + l16) * (3 * DD) + DD + hh * HDD;
#pragma unroll
        for (int j = 0; j < 4; ++j) {
            int kk = 4 * j + half * 2;
            v2f bf; bf.x = kp[kk]; bf.y = kp[kk + 1];
            sc = __builtin_amdgcn_wmma_f32_16x16x4_f32(false, qa[j], false, bf, (short)0,
                                                       sc, false, false);
        }
#pragma unroll
        for (int r = 0; r < 8; ++r) {
            float v = sc[r] * scale;
            float mx = v;
            mx = fmaxf(mx, __shfl_xor(mx, 8, 32));
            mx = fmaxf(mx, __shfl_xor(mx, 4, 32));
            mx = fmaxf(mx, __shfl_xor(mx, 2, 32));
            mx = fmaxf(mx, __shfl_xor(mx, 1, 32));
            float mnew = fmaxf(mrow[r], mx);
            float p = __expf(v - mnew);
            float ps = p;
            ps += __shfl_xor(ps, 8, 32);
            ps += __shfl_xor(ps, 4, 32);
            ps += __shfl_xor(ps, 2, 32);
            ps += __shfl_xor(ps, 1, 32);
            float cf = __expf(mrow[r] - mnew);
            srow[r] = srow[r] * cf + ps;
            mrow[r] = mnew;
            o[r] = o[r] * cf;
            psh[w][r + half * 8][l16] = p;   // row-major P tile
        }
        __syncthreads();
        const float* vp = qkv + (size_t)kbase * (3 * DD) + 2 * DD + hh * HDD + l16;
#pragma unroll
        for (int j = 0; j < 4; ++j) {
            int kk = 4 * j + half * 2;
            v2f pa; pa.x = psh[w][l16][kk]; pa.y = psh[w][l16][kk + 1];
            v2f vb; vb.x = vp[(size_t)kk * (3 * DD)]; vb.y = vp[(size_t)(kk + 1) * (3 * DD)];
            o = __builtin_amdgcn_wmma_f32_16x16x4_f32(false, pa, false, vb, (short)0, o,
                                                      false, false);
        }
        __syncthreads();
    }
#pragma unroll
    for (int r = 0; r < 8; ++r) {
        int row = s0 + r + half * 8;
        o_out[(size_t)row * DD + hh * HDD + l16] = o[r] / srow[r];
    }
}

// ------------- mean pool over S -------------
__global__ void k_pool(const float* __restrict__ h, float* __restrict__ g) {
    int d = threadIdx.x;
    float acc = 0.f;
    for (int s = 0; s < SS; ++s) acc += h[(size_t)s * DD + d];
    g[d] = acc * (1.f / SS);
}

// ------------- MLP heads -------------
__global__ void k_heads(const float* __restrict__ g,
                        const float* rW1, const float* rb1, const float* rW2, const float* rb2,
                        const float* mW1, const float* mb1, const float* mW2, const float* mb2,
                        const float* uW1, const float* ub1, const float* uW2, const float* ub2,
                        float* out) {
    __shared__ float gs[DD];
    __shared__ float hid[64];
    int tid = threadIdx.x;
    gs[tid] = g[tid];
    __syncthreads();
    if (tid < 64) {
        float a = rb1[tid];
        for (int d = 0; d < DD; ++d) a = fmaf(gs[d], rW1[d * 64 + tid], a);
        hid[tid] = fmaxf(a, 0.f);
    }
    __syncthreads();
    if (tid == 0) {
        float a = rb2[0];
        for (int j = 0; j < 64; ++j) a = fmaf(hid[j], rW2[j], a);
        out[0] = 1.f / (1.f + __expf(-a));
    }
    __syncthreads();
    if (tid < 32) {
        float a = mb1[tid];
        for (int d = 0; d < DD; ++d) a = fmaf(gs[d], mW1[d * 32 + tid], a);
        hid[tid] = fmaxf(a, 0.f);
    }
    __syncthreads();
    if (tid < 8) {
        float a = mb2[tid];
        for (int j = 0; j < 32; ++j) a = fmaf(hid[j], mW2[j * 8 + tid], a);
        out[1 + tid] = a;
    }
    __syncthreads();
    if (tid < 32) {
        float a = ub1[tid];
        for (int d = 0; d < DD; ++d) a = fmaf(gs[d], uW1[d * 32 + tid], a);
        hid[tid] = fmaxf(a, 0.f);
    }
    __syncthreads();
    if (tid < 2) {
        float a = ub2[tid];
        for (int j = 0; j < 32; ++j) a = fmaf(hid[j], uW2[j * 2 + tid], a);
        out[9 + tid] = (a > 20.f) ? a : log1pf(__expf(a));
    }
}

extern "C" void kernel_launch(void* const* d_in, const int* in_sizes, int n_in,
                              void* d_out, int out_size, void* d_ws, size_t ws_size,
                              hipStream_t stream) {
    const float* x        = (const float*)d_in[0];
    const int*   ei       = (const int*)d_in[1];
    const float* gat_W    = (const float*)d_in[2];
    const float* gat_asrc = (const float*)d_in[3];
    const float* gat_adst = (const float*)d_in[4];
    const float* gat_b    = (const float*)d_in[5];
    const float* ln_g     = (const float*)d_in[6];
    const float* ln_b     = (const float*)d_in[7];
    const float* Wqkv     = (const float*)d_in[8];
    const float* bqkv     = (const float*)d_in[9];
    const float* Wo       = (const float*)d_in[10];
    const float* bo       = (const float*)d_in[11];
    const float* ln1g     = (const float*)d_in[12];
    const float* ln1b     = (const float*)d_in[13];
    const float* W1       = (const float*)d_in[14];
    const float* b1       = (const float*)d_in[15];
    const float* W2       = (const float*)d_in[16];
    const float* b2       = (const float*)d_in[17];
    const float* ln2g     = (const float*)d_in[18];
    const float* ln2b     = (const float*)d_in[19];
    const float* rW1 = (const float*)d_in[20]; const float* rb1 = (const float*)d_in[21];
    const float* rW2 = (const float*)d_in[22]; const float* rb2 = (const float*)d_in[23];
    const float* mW1 = (const float*)d_in[24]; const float* mb1 = (const float*)d_in[25];
    const float* mW2 = (const float*)d_in[26]; const float* mb2 = (const float*)d_in[27];
    const float* uW1 = (const float*)d_in[28]; const float* ub1 = (const float*)d_in[29];
    const float* uW2 = (const float*)d_in[30]; const float* ub2 = (const float*)d_in[31];
    float* out = (float*)d_out;

    int E  = in_sizes[1] / 2;
    int EN = E + NN;

    float* ws = (float*)d_ws;
    size_t off = 0;
    float* hgat  = ws + off; off += (size_t)TT * NN * HH * DD;
    float* as_s  = ws + off; off += (size_t)TT * NN * HH;
    float* ad_s  = ws + off; off += (size_t)TT * NN * HH;
    unsigned* mbuf = (unsigned*)(ws + off); off += (size_t)TT * NN * HH;
    float* ssum  = ws + off; off += (size_t)TT * NN * HH;
    float* exbuf = ws + off; off += (size_t)TT * EN * HH;
    float* gout  = ws + off; off += (size_t)TT * NN * DD;
    float* h0    = ws + off; off += (size_t)SS * DD;
    float* qkv   = ws + off; off += (size_t)SS * 3 * DD;
    float* atto  = ws + off; off += (size_t)SS * DD;
    float* tmp   = ws + off; off += (size_t)SS * DD;
    float* ffn   = ws + off; off += (size_t)SS * FFN;
    float* gvec  = ws + off; off += DD;

    hipMemsetAsync(gout, 0, (size_t)TT * NN * DD * sizeof(float), stream);
    hipMemsetAsync(mbuf, 0, (size_t)TT * NN * HH * sizeof(unsigned), stream);
    hipMemsetAsync(ssum, 0, (size_t)TT * NN * HH * sizeof(float), stream);

    // ---- GAT ----
    k_gat_lin<<<(TT * NN * HH * DD + 255) / 256, 256, 0, stream>>>(x, gat_W, hgat);
    k_gat_alpha<<<(TT * NN * HH + 7) / 8, 256, 0, stream>>>(hgat, gat_asrc, gat_adst, as_s, ad_s);
    int nE = TT * EN;
    k_edge_score<<<(nE + 255) / 256, 256, 0, stream>>>(ei, as_s, ad_s, exbuf, mbuf, E);
    k_edge_exp<<<(nE + 255) / 256, 256, 0, stream>>>(ei, mbuf, exbuf, ssum, E);
    k_edge_agg<<<nE, DD, 0, stream>>>(ei, exbuf, ssum, hgat, gout, E);
    k_ln<<<SS, DD, 0, stream>>>(gout, gat_b, 1, ln_g, ln_b, h0);

    // ---- Transformer encoder, 3 layers (post-norm, ReLU FFN) ----
    for (int l = 0; l < LLAY; ++l) {
        k_gemm<<<(SS / 64) * ((3 * DD) / 64), 256, 0, stream>>>(
            h0, Wqkv + (size_t)l * DD * 3 * DD, bqkv + (size_t)l * 3 * DD, qkv,
            SS, 3 * DD, DD, 0);
        k_attn<<<(HH * (SS / 16)) / 4, 128, 0, stream>>>(qkv, atto);
        k_gemm<<<(SS / 64) * (DD / 64), 256, 0, stream>>>(
            atto, Wo + (size_t)l * DD * DD, bo + (size_t)l * DD, tmp, SS, DD, DD, 0);
        k_ln<<<SS, DD, 0, stream>>>(h0, tmp, 0, ln1g + (size_t)l * DD, ln1b + (size_t)l * DD, h0);
        k_gemm<<<(SS / 64) * (FFN / 64), 256, 0, stream>>>(
            h0, W1 + (size_t)l * DD * FFN, b1 + (size_t)l * FFN, ffn, SS, FFN, DD, 1);
        k_gemm<<<(SS / 64) * (DD / 64), 256, 0, stream>>>(
            ffn, W2 + (size_t)l * FFN * DD, b2 + (size_t)l * DD, tmp, SS, DD, FFN, 0);
        k_ln<<<SS, DD, 0, stream>>>(h0, tmp, 0, ln2g + (size_t)l * DD, ln2b + (size_t)l * DD, h0);
    }

    // ---- pool + heads ----
    k_pool<<<1, DD, 0, stream>>>(h0, gvec);
    k_heads<<<1, 128, 0, stream>>>(gvec, rW1, rb1, rW2, rb2, mW1, mb1, mW2, mb2,
                                   uW1, ub1, uW2, ub2, out);
}